// SECLoss_24300924961223
// MI455X (gfx1250) — compile-verified
//
#include <hip/hip_runtime.h>
#include <math.h>

// SEC loss for MI455X (gfx1250, wave32).
// Memory-bound elementwise passes + 21504 LDS bitonic sorts for GWRP pooling.
// Cross-lane sum reductions implemented with V_WMMA_F32_16X16X4_F32 (D = A x ones).

#define NCH   21
#define NFG   20
#define HW    1681
#define SORT_N 2048
#define MIN_PROB_F 1e-4f
#define Q_FG 0.996f
#define Q_BG 0.999f

typedef float v2f __attribute__((ext_vector_type(2)));
typedef float v8f __attribute__((ext_vector_type(8)));

// Wave32 sum reduction via fp32 WMMA: A[m,k] holds lane partials (K=0 from lanes
// 0-15, K=2 from lanes 16-31 per the 16x4 f32 A layout), B = ones =>
// D[m,n] = p_m + p_{m+16}. Lane L<16 holds D[0..7][L] in its 8 VGPRs, lane L>=16
// holds D[8..15][L-16]; hsum + shfl_xor(16) yields the full 32-lane sum in all lanes.
__device__ __forceinline__ float wave_sum_wmma(float x) {
  v2f a = { x, 0.0f };
  v2f b = { 1.0f, 1.0f };
  v8f c = {};
  c = __builtin_amdgcn_wmma_f32_16x16x4_f32(false, a, false, b, (short)0, c, false, false);
  float s = c[0] + c[1] + c[2] + c[3] + c[4] + c[5] + c[6] + c[7];
  s += __shfl_xor(s, 16, 32);
  return s;
}

// Block (256 threads = 8 waves) sum; result valid in all threads.
__device__ __forceinline__ float block_sum(float x, float* red) {
  float w = wave_sum_wmma(x);
  __syncthreads();                       // protect red[] reuse across calls
  if ((threadIdx.x & 31u) == 0u) red[threadIdx.x >> 5] = w;
  __syncthreads();
  float t = 0.0f;
#pragma unroll
  for (int i = 0; i < 8; ++i) t += red[i];
  return t;
}

__global__ void k_init(float* acc) {
  if (threadIdx.x < 4) acc[threadIdx.x] = 0.0f;
}

// One block per image n: softmax stats (cached to ws), loss_s numerator/count,
// loss_c partial sum.
__global__ void __launch_bounds__(256) k_main(
    const float* __restrict__ out, const float* __restrict__ gt,
    const float* __restrict__ crf, float* __restrict__ wm, float* __restrict__ wsum,
    float* __restrict__ Sn, float* __restrict__ Cn, float* __restrict__ acc) {
  __shared__ float red[8];
  const int n = blockIdx.x;
  const float invZ = 1.0f / (1.0f + (float)NCH * MIN_PROB_F);
  const size_t base  = (size_t)n * NCH * HW;
  const size_t nbase = (size_t)n * HW;

  float sacc = 0.0f, cacc = 0.0f, lcacc = 0.0f;
  for (int p = threadIdx.x; p < HW; p += 256) {
    float x[NCH];
    float m = -3.402823466e38f;
#pragma unroll
    for (int c = 0; c < NCH; ++c) {
      x[c] = out[base + (size_t)c * HW + p];
      m = fmaxf(m, x[c]);
    }
    float se = 0.0f;
#pragma unroll
    for (int c = 0; c < NCH; ++c) se += expf(x[c] - m);
    wm[nbase + p]   = m;
    wsum[nbase + p] = se;
    const float inv_se = 1.0f / se;
#pragma unroll
    for (int c = 0; c < NCH; ++c) {
      const float prob = (expf(x[c] - m) * inv_se + MIN_PROB_F) * invZ;
      const float lp = logf(prob);
      const float g = gt[base + (size_t)c * HW + p];
      sacc = fmaf(g, lp, sacc);
      cacc += g;
      const float cl = crf[base + (size_t)c * HW + p];
      lcacc = fmaf(expf(cl), cl - lp, lcacc);
    }
  }
  const float st = block_sum(sacc, red);
  const float ct = block_sum(cacc, red);
  const float lt = block_sum(lcacc, red);
  if (threadIdx.x == 0) {
    Sn[n] = st;
    Cn[n] = ct;
    atomicAdd(&acc[0], lt);              // global sum for loss_c
  }
}

// One block per (n,c): bitonic sort 2048 in LDS, GWRP weighted mean, channel max,
// label-masked contribution to loss_1/2/3.
__global__ void __launch_bounds__(256) k_sort(
    const float* __restrict__ out, const float* __restrict__ label,
    const float* __restrict__ wm, const float* __restrict__ wsum,
    float* __restrict__ acc, int Nimg) {
  __shared__ float buf[SORT_N];
  __shared__ float red[8];
  const int n = blockIdx.x / NCH;
  const int c = blockIdx.x % NCH;
  const float invZ = 1.0f / (1.0f + (float)NCH * MIN_PROB_F);
  const size_t cbase = ((size_t)n * NCH + c) * HW;
  const size_t nbase = (size_t)n * HW;

  for (int i = threadIdx.x; i < SORT_N; i += 256) {
    float v = 3.402823466e38f;           // pads sort above all real probs
    if (i < HW) {
      const float x = out[cbase + i];
      v = (expf(x - wm[nbase + i]) / wsum[nbase + i] + MIN_PROB_F) * invZ;
    }
    buf[i] = v;
  }
  __syncthreads();

  // Ascending bitonic sort of 2048 elements, 256 threads.
  for (unsigned k = 2; k <= SORT_N; k <<= 1) {
    for (unsigned j = k >> 1; j > 0; j >>= 1) {
      for (unsigned i = threadIdx.x; i < SORT_N; i += 256) {
        const unsigned ixj = i ^ j;
        if (ixj > i) {
          const float a = buf[i];
          const float b = buf[ixj];
          const bool up = ((i & k) == 0);
          if ((a > b) == up) { buf[i] = b; buf[ixj] = a; }
        }
      }
      __syncthreads();
    }
  }

  const float q   = (c == 0) ? Q_BG : Q_FG;
  const float l2q = log2f(q);
  float part = 0.0f;
  for (int i = threadIdx.x; i < HW; i += 256)
    part = fmaf(buf[i], exp2f((float)(HW - 1 - i) * l2q), part);
  const float tot = block_sum(part, red);

  if (threadIdx.x == 0) {
    const float Z    = (1.0f - exp2f((float)HW * l2q)) / (1.0f - q);
    const float mean = tot / Z;
    const float invN = 1.0f / (float)Nimg;
    float contrib;
    if (c == 0) {                         // loss_3
      contrib = -logf(mean) * invN;
    } else {
      float sum_stat = 0.0f;
      for (int cc = 1; cc < NCH; ++cc)
        sum_stat += (label[n * NCH + cc] > 0.5f) ? 1.0f : 0.0f;
      const bool stat = label[n * NCH + c] > 0.5f;
      if (stat) {                         // loss_1
        contrib = -logf(mean) / sum_stat * invN;
      } else {                            // loss_2 (uses spatial max)
        const float pmax = buf[HW - 1];
        contrib = -logf(1.0f - pmax) / ((float)NFG - sum_stat) * invN;
      }
    }
    atomicAdd(&acc[1], contrib);
  }
}

__global__ void __launch_bounds__(256) k_final(
    const float* __restrict__ Sn, const float* __restrict__ Cn,
    const float* __restrict__ acc, float* __restrict__ outp, int Nimg) {
  __shared__ float red[8];
  float part = 0.0f;
  for (int n = threadIdx.x; n < Nimg; n += 256) part += -Sn[n] / Cn[n];
  const float tot = block_sum(part, red);
  if (threadIdx.x == 0) {
    const float loss_s = tot / (float)Nimg;
    const float loss_c = acc[0] / ((float)Nimg * (float)HW);
    outp[0] = loss_s + acc[1] + loss_c;
  }
}

extern "C" void kernel_launch(void* const* d_in, const int* in_sizes, int n_in,
                              void* d_out, int out_size, void* d_ws, size_t ws_size,
                              hipStream_t stream) {
  const float* output = (const float*)d_in[0];
  const float* gt_cue = (const float*)d_in[1];
  const float* label  = (const float*)d_in[2];
  const float* crf    = (const float*)d_in[3];
  const int Nimg = in_sizes[0] / (NCH * HW);

  float* wsf = (float*)d_ws;
  const size_t NHW = (size_t)Nimg * HW;
  float* wm   = wsf;                 // per-pixel channel max
  float* wsum = wsf + NHW;           // per-pixel sum(exp(x-max))
  float* Sn   = wsf + 2 * NHW;       // per-image sum(gt*logp)
  float* Cn   = Sn + Nimg;           // per-image sum(gt)
  float* acc  = Cn + Nimg;           // [0]=loss_c sum, [1]=loss_e sum

  k_init <<<1, 64, 0, stream>>>(acc);
  k_main <<<Nimg, 256, 0, stream>>>(output, gt_cue, crf, wm, wsum, Sn, Cn, acc);
  k_sort <<<Nimg * NCH, 256, 0, stream>>>(output, label, wm, wsum, acc, Nimg);
  k_final<<<1, 256, 0, stream>>>(Sn, Cn, acc, (float*)d_out, Nimg);
}